// ImprovedSNL_7155415515165
// MI455X (gfx1250) — compile-verified
//
#include <hip/hip_runtime.h>
#include <hip/hip_bf16.h>

#define NPIX 4096
#define NB 4
#define CCH 64
#define TCH 32
#define EPSBN 1e-5f

typedef float v2f __attribute__((ext_vector_type(2)));
typedef float v8f __attribute__((ext_vector_type(8)));

// D = A(16x4) * B(4x16) + C, f32, wave32.
// A frag: lane l: a0=A[l&15][(l>>4)*2], a1=A[l&15][(l>>4)*2+1]
// B frag: lane l: b0=B[(l>>4)*2][l&15], b1=B[(l>>4)*2+1][l&15]
// D: lane l, reg r -> M = r + 8*(l>=16), N = l&15
__device__ __forceinline__ v8f wmma4(v2f a, v2f b, v8f c) {
  return __builtin_amdgcn_wmma_f32_16x16x4_f32(false, a, false, b, (short)0, c,
                                               false, false);
}

// ---------------- projection: tpcr[b] = act([Wt;Wp;Wcol;Wrow] @ xf[b]) -------
// rows 0-31: t=relu(Wt x), 32-63: p=relu(Wp x), 64-95: cf=Wcol x, 96-127: rf=Wrow x
__global__ void k_proj(const float* __restrict__ xf,
                       const float* __restrict__ Wt, const float* __restrict__ Wp,
                       const float* __restrict__ Wc, const float* __restrict__ Wr,
                       float* __restrict__ tpcr) {
  int wave = (blockIdx.x * blockDim.x + threadIdx.x) >> 5;
  int lane = threadIdx.x & 31;
  int nt = wave & 255;          // 256 n-tiles of 16
  int mt = (wave >> 8) & 7;     // 8 m-tiles of 16
  int b  = wave >> 11;          // 4 batches
  const float* xfb = xf + b * CCH * NPIX;
  float* outb = tpcr + b * 128 * NPIX;
  int row = mt * 16 + (lane & 15);
  int col = nt * 16 + (lane & 15);
  int klo = (lane >> 4) * 2;
  const float* Wsrc; int rloc;          // uniform per wave (16-row blocks)
  if (row < 32)      { Wsrc = Wt; rloc = row; }
  else if (row < 64) { Wsrc = Wp; rloc = row - 32; }
  else if (row < 96) { Wsrc = Wc; rloc = row - 64; }
  else               { Wsrc = Wr; rloc = row - 96; }
  v8f acc = {};
#pragma unroll
  for (int k0 = 0; k0 < CCH; k0 += 4) {
    v2f a, bb;
    a.x = Wsrc[rloc * CCH + k0 + klo];
    a.y = Wsrc[rloc * CCH + k0 + klo + 1];
    bb.x = xfb[(k0 + klo) * NPIX + col];
    bb.y = xfb[(k0 + klo + 1) * NPIX + col];
    acc = wmma4(a, bb, acc);
  }
  bool do_relu = (mt < 4);              // uniform
#pragma unroll
  for (int r = 0; r < 8; ++r) {
    int m = mt * 16 + r + ((lane >> 4) << 3);
    float v = acc[r];
    if (do_relu) v = fmaxf(v, 0.0f);
    outb[m * NPIX + col] = v;
  }
}

// ---------------- row sums of t,p -> svec[b][0..31]=s_t, [32..63]=s_p -------
__global__ void k_rowsum(const float* __restrict__ tpcr, float* __restrict__ svec) {
  int b = blockIdx.x >> 6;
  int c = blockIdx.x & 63;
  const float* rowp = tpcr + (b * 128 + c) * NPIX;
  float s = 0.f;
  for (int n = threadIdx.x; n < NPIX; n += 256) s += rowp[n];
  __shared__ float red[256];
  red[threadIdx.x] = s;
  __syncthreads();
  for (int off = 128; off > 0; off >>= 1) {
    if ((int)threadIdx.x < off) red[threadIdx.x] += red[threadIdx.x + off];
    __syncthreads();
  }
  if (threadIdx.x == 0) svec[b * 64 + c] = red[0];
}

// ---------------- deg[n]=0.5*(t[:,n].s_p + p[:,n].s_t); dinv = rsqrt --------
__global__ void k_deg(const float* __restrict__ tpcr, const float* __restrict__ svec,
                      float* __restrict__ dinv) {
  int idx = blockIdx.x * blockDim.x + threadIdx.x;   // [0, 4*4096)
  int b = idx >> 12;
  int n = idx & (NPIX - 1);
  const float* tb = tpcr + b * 128 * NPIX;
  const float* sb = svec + b * 64;
  float deg = 0.f;
#pragma unroll 8
  for (int c = 0; c < TCH; ++c) {
    deg += tb[c * NPIX + n] * sb[32 + c];
    deg += tb[(32 + c) * NPIX + n] * sb[c];
  }
  deg *= 0.5f;
  dinv[idx] = (deg != 0.0f) ? rsqrtf(deg) : 0.0f;
}

__global__ void k_zero(float* __restrict__ p, int n) {
  int i = blockIdx.x * blockDim.x + threadIdx.x;
  if (i < n) p[i] = 0.f;
}

// ---------------- Gram mats (32x32, K=4096, split-K=4, atomic f32) ----------
// mat0: G1=(cf*dinv)@t^T  mat1: G2=(cf*dinv)@p^T
// mat2: H1=(rf*dinv)@p^T  mat3: H2=(rf*dinv)@t^T
__global__ void k_gmat(const float* __restrict__ tpcr, const float* __restrict__ dinv,
                       float* __restrict__ G) {
  int wave = (blockIdx.x * blockDim.x + threadIdx.x) >> 5;
  int lane = threadIdx.x & 31;
  int kc  = wave & 3;
  int tj  = (wave >> 2) & 1;
  int ti  = (wave >> 3) & 1;
  int mat = (wave >> 4) & 3;
  int b   = wave >> 6;
  int Abase = (mat < 2) ? 64 : 96;                 // cf : rf
  int Bbase = (mat == 0 || mat == 3) ? 0 : 32;     // t : p
  const float* tb = tpcr + b * 128 * NPIX;
  const float* dv = dinv + b * NPIX;
  int arow = Abase + ti * 16 + (lane & 15);
  int brow = Bbase + tj * 16 + (lane & 15);
  int klo = (lane >> 4) * 2;
  int n0 = kc * 1024;
  v8f acc = {};
  for (int kk = 0; kk < 1024; kk += 4) {
    int n = n0 + kk + klo;
    v2f a, bb;
    a.x = tb[arow * NPIX + n] * dv[n];
    a.y = tb[arow * NPIX + n + 1] * dv[n + 1];
    bb.x = tb[brow * NPIX + n];
    bb.y = tb[brow * NPIX + n + 1];
    acc = wmma4(a, bb, acc);
  }
  float* Gm = G + (b * 4 + mat) * 32 * 32;
#pragma unroll
  for (int r = 0; r < 8; ++r) {
    int m = ti * 16 + r + ((lane >> 4) << 3);
    int j = tj * 16 + (lane & 15);
    atomicAdd(&Gm[m * 32 + j], acc[r]);
  }
}

// ---------------- fused mid + out --------------------------------------------
// Per (b, 16-col strip): 4 waves each build one 16x16 tile of
//   uv = [ 0.5*dinv*(G1@p + G2@t) ; 0.5*dinv*(H1@t + H2@p) ]   (64x16)
// staged in LDS, then each wave computes 16 rows of
//   out_pre = [W1 | W2] @ uv                                    (64x16)
// with B-fragments served from LDS.
__global__ void k_midout(const float* __restrict__ tpcr, const float* __restrict__ G,
                         const float* __restrict__ dinv,
                         const float* __restrict__ W1, const float* __restrict__ W2,
                         float* __restrict__ outp) {
  __shared__ float uvt[64 * 16];                    // 4 KB tile, row-major [m][c]
  int w    = threadIdx.x >> 5;                      // wave id = mt for phase 1
  int lane = threadIdx.x & 31;
  int nt = blockIdx.x & 255;
  int b  = blockIdx.x >> 8;
  const float* tb = tpcr + b * 128 * NPIX;
  const float* dv = dinv + b * NPIX;
  int col  = nt * 16 + (lane & 15);
  int klo  = (lane >> 4) * 2;
  int lcol = lane & 15;

  // ---- phase 1: uv tile ----
  {
    int mt = w;
    bool isv = (mt >= 2);                           // uniform per wave
    int ti = mt & 1;
    const float* Ga = G + (b * 4 + (isv ? 2 : 0)) * 1024;
    const float* Gb = G + (b * 4 + (isv ? 3 : 1)) * 1024;
    int br0 = isv ? 0 : 32;                         // H1 pairs t; G1 pairs p
    int br1 = isv ? 32 : 0;                         // H2 pairs p; G2 pairs t
    int arow = ti * 16 + (lane & 15);
    v8f acc = {};
#pragma unroll
    for (int k0 = 0; k0 < 32; k0 += 4) {
      v2f a, bb;
      a.x = Ga[arow * 32 + k0 + klo];
      a.y = Ga[arow * 32 + k0 + klo + 1];
      bb.x = tb[(br0 + k0 + klo) * NPIX + col];
      bb.y = tb[(br0 + k0 + klo + 1) * NPIX + col];
      acc = wmma4(a, bb, acc);
    }
#pragma unroll
    for (int k0 = 0; k0 < 32; k0 += 4) {
      v2f a, bb;
      a.x = Gb[arow * 32 + k0 + klo];
      a.y = Gb[arow * 32 + k0 + klo + 1];
      bb.x = tb[(br1 + k0 + klo) * NPIX + col];
      bb.y = tb[(br1 + k0 + klo + 1) * NPIX + col];
      acc = wmma4(a, bb, acc);
    }
    float scale = 0.5f * dv[col];
#pragma unroll
    for (int r = 0; r < 8; ++r) {
      int m = mt * 16 + r + ((lane >> 4) << 3);
      uvt[m * 16 + lcol] = acc[r] * scale;
    }
  }
  __syncthreads();

  // ---- phase 2: out rows [w*16, w*16+16) from LDS tile ----
  {
    int row = w * 16 + (lane & 15);
    v8f acc = {};
#pragma unroll
    for (int k0 = 0; k0 < 64; k0 += 4) {
      int k = k0 + klo;
      const float* Wsrc = (k0 < 32) ? W1 : W2;      // uniform
      int kk = (k0 < 32) ? k : (k - 32);
      v2f a, bb;
      a.x = Wsrc[row * 32 + kk];
      a.y = Wsrc[row * 32 + kk + 1];
      bb.x = uvt[k * 16 + lcol];                    // ds_load, conflict-free
      bb.y = uvt[(k + 1) * 16 + lcol];
      acc = wmma4(a, bb, acc);
    }
    float* ob = outp + b * 64 * NPIX;
#pragma unroll
    for (int r = 0; r < 8; ++r) {
      int m = w * 16 + r + ((lane >> 4) << 3);
      ob[m * NPIX + col] = acc[r];
    }
  }
}

// ---------------- BN stats per channel over (B, N) --------------------------
__global__ void k_bnstats(const float* __restrict__ outp, float* __restrict__ stats) {
  int c = blockIdx.x;
  float s = 0.f, s2 = 0.f;
  for (int i = threadIdx.x; i < NB * NPIX; i += 256) {
    int b = i >> 12;
    int n = i & (NPIX - 1);
    float v = outp[(b * 64 + c) * NPIX + n];
    s += v; s2 += v * v;
  }
  __shared__ float r1[256], r2[256];
  r1[threadIdx.x] = s; r2[threadIdx.x] = s2;
  __syncthreads();
  for (int off = 128; off > 0; off >>= 1) {
    if ((int)threadIdx.x < off) {
      r1[threadIdx.x] += r1[threadIdx.x + off];
      r2[threadIdx.x] += r2[threadIdx.x + off];
    }
    __syncthreads();
  }
  if (threadIdx.x == 0) {
    const float inv = 1.0f / (float)(NB * NPIX);
    float mean = r1[0] * inv;
    stats[c] = mean;
    stats[64 + c] = r2[0] * inv - mean * mean;   // biased var
  }
}

// ---------------- BN apply + residual ---------------------------------------
__global__ void k_bnapply(const float* __restrict__ outp, const float* __restrict__ stats,
                          const float* __restrict__ gamma, const float* __restrict__ beta,
                          const float* __restrict__ xf_in, float* __restrict__ xf_out) {
  int idx = blockIdx.x * blockDim.x + threadIdx.x;   // 4*64*4096
  int c = (idx >> 12) & 63;
  float mean = stats[c];
  float g = gamma[c] * rsqrtf(stats[64 + c] + EPSBN);
  xf_out[idx] = g * (outp[idx] - mean) + beta[c] + xf_in[idx];
}

extern "C" void kernel_launch(void* const* d_in, const int* in_sizes, int n_in,
                              void* d_out, int out_size, void* d_ws, size_t ws_size,
                              hipStream_t stream) {
  (void)in_sizes; (void)n_in; (void)out_size; (void)ws_size;
  const float* x     = (const float*)d_in[0];
  const float* Wt    = (const float*)d_in[1];
  const float* Wp    = (const float*)d_in[2];
  const float* Wrow  = (const float*)d_in[3];
  const float* Wcol  = (const float*)d_in[4];
  const float* W1    = (const float*)d_in[5];
  const float* W2    = (const float*)d_in[6];
  const float* gamma = (const float*)d_in[7];
  const float* beta  = (const float*)d_in[8];

  float* w      = (float*)d_ws;
  float* xf_buf = w;                    // 4*64*4096   = 1048576 f
  float* tpcr   = xf_buf + 1048576;     // 4*128*4096  = 2097152 f
  float* svec   = tpcr + 2097152;       // 4*64        = 256 f
  float* dinv   = svec + 256;           // 4*4096      = 16384 f
  float* G      = dinv + 16384;         // 4*4*32*32   = 16384 f
  float* outp   = G + 16384;            // 4*64*4096   = 1048576 f
  float* stats  = outp + 1048576;       // 128 f        (total ~16.1 MB)

  for (int stage = 0; stage < 2; ++stage) {
    const float* xf_in = (stage == 0) ? x : xf_buf;
    float* xf_out = (stage == 0) ? xf_buf : (float*)d_out;

    k_zero   <<<64,   256, 0, stream>>>(G, 16384);
    k_proj   <<<2048, 128, 0, stream>>>(xf_in, Wt, Wp, Wcol, Wrow, tpcr);
    k_rowsum <<<256,  256, 0, stream>>>(tpcr, svec);
    k_deg    <<<64,   256, 0, stream>>>(tpcr, svec, dinv);
    k_gmat   <<<64,   128, 0, stream>>>(tpcr, dinv, G);
    k_midout <<<1024, 128, 0, stream>>>(tpcr, G, dinv, W1, W2, outp);
    k_bnstats<<<64,   256, 0, stream>>>(outp, stats);
    k_bnapply<<<4096, 256, 0, stream>>>(outp, stats, gamma, beta, xf_in, xf_out);
  }
}